// ApproximationLayer_24232205484308
// MI455X (gfx1250) — compile-verified
//
#include <hip/hip_runtime.h>

// Native 4-wide float vector so __builtin_nontemporal_{load,store} accept it
typedef float v4f __attribute__((ext_vector_type(4)));

// _mute(x): frexp convention x = m*2^e, m in [0.5,1). Fires only when e>1,
// i.e. |x| >= 2 and finite; result is exactly m with sign preserved ->
// rewrite biased exponent field to 126. Identity otherwise (0, denorm,
// |x|<2, inf, NaN: frexp returns e<=1 / e==0 for all of those).
__device__ __forceinline__ float mute_sel(float x, unsigned apply) {
  unsigned u = __float_as_uint(x);
  unsigned e = (u >> 23) & 0xFFu;
  bool fire = (apply != 0u) & (e > 127u) & (e < 255u);
  unsigned muted = (u & 0x807FFFFFu) | 0x3F000000u;  // exponent field = 126
  return fire ? __uint_as_float(muted) : x;
}

__device__ __forceinline__ void build_masks(const int* __restrict__ rows,
                                            const int* __restrict__ cols,
                                            unsigned& rowmask, unsigned& colmask) {
  rowmask = 0u; colmask = 0u;
#pragma unroll
  for (int k = 0; k < 4; ++k) {
    rowmask |= 1u << (unsigned)rows[k];
    colmask |= 1u << (unsigned)cols[k];
  }
}

__device__ __forceinline__ v4f apply_mask(v4f v, unsigned m) {
  v[0] = mute_sel(v[0], m & 1u);
  v[1] = mute_sel(v[1], m & 2u);
  v[2] = mute_sel(v[2], m & 4u);
  v[3] = mute_sel(v[3], m & 8u);
  return v;
}

// 784 floats per 28x28 image = 196 float4s; 7 float4s per row of 28.
// 28 % 4 == 0 -> a float4 never crosses a row (or image) boundary.
__device__ __forceinline__ unsigned mute_mask_for(unsigned i, unsigned rowmask,
                                                  unsigned colmask) {
  unsigned q   = i % 196u;            // float4 index within one image
  unsigned row = q / 7u;              // uniform row for all 4 floats
  unsigned c0  = (q - row * 7u) * 4u; // first column covered by this vector
  return ((rowmask >> row) & 1u) ? 0xFu : ((colmask >> c0) & 0xFu);
}

// Dual-vector kernel: half % 196 == 0, so vectors i and i+half occupy the
// same (row, col) position in their images -> one mask computation serves
// two 128-bit load/store pairs, halving instruction overhead per byte.
__global__ __launch_bounds__(256) void mute_rc_x2_kernel(
    const v4f* __restrict__ x,
    const int* __restrict__ rows,
    const int* __restrict__ cols,
    v4f* __restrict__ out,
    unsigned half)
{
  unsigned i = blockIdx.x * 256u + threadIdx.x;
  if (i >= half) return;

  unsigned rowmask, colmask;
  build_masks(rows, cols, rowmask, colmask);
  unsigned m = mute_mask_for(i, rowmask, colmask);

  // NT hints: zero-reuse 822 MB stream -> keep it out of L2/WGP$.
  v4f a = __builtin_nontemporal_load(&x[i]);
  v4f b = __builtin_nontemporal_load(&x[i + half]);
  a = apply_mask(a, m);
  b = apply_mask(b, m);
  __builtin_nontemporal_store(a, &out[i]);
  __builtin_nontemporal_store(b, &out[i + half]);
}

// Generic fallback: one float4 per thread.
__global__ __launch_bounds__(256) void mute_rc_kernel(
    const v4f* __restrict__ x,
    const int* __restrict__ rows,
    const int* __restrict__ cols,
    v4f* __restrict__ out,
    unsigned nvec)
{
  unsigned i = blockIdx.x * 256u + threadIdx.x;
  if (i >= nvec) return;

  unsigned rowmask, colmask;
  build_masks(rows, cols, rowmask, colmask);
  unsigned m = mute_mask_for(i, rowmask, colmask);

  v4f v = __builtin_nontemporal_load(&x[i]);
  v = apply_mask(v, m);
  __builtin_nontemporal_store(v, &out[i]);
}

extern "C" void kernel_launch(void* const* d_in, const int* in_sizes, int n_in,
                              void* d_out, int out_size, void* d_ws, size_t ws_size,
                              hipStream_t stream) {
  const v4f* x    = (const v4f*)d_in[0];
  const int* rows = (const int*)d_in[1];
  const int* cols = (const int*)d_in[2];
  v4f* out        = (v4f*)d_out;

  unsigned nvec = (unsigned)in_sizes[0] / 4u;  // 131072*28*28/4 = 25,690,112
  unsigned half = nvec / 2u;                   // 12,845,056 = 196 * 65,536

  if ((nvec % 2u == 0u) && (half % 196u == 0u)) {
    unsigned blocks = (half + 255u) / 256u;    // 50,176 blocks of 8 wave32s
    mute_rc_x2_kernel<<<blocks, 256, 0, stream>>>(x, rows, cols, out, half);
  } else {
    unsigned blocks = (nvec + 255u) / 256u;
    mute_rc_kernel<<<blocks, 256, 0, stream>>>(x, rows, cols, out, nvec);
  }
}